// AffineLayer_Numpy_53386443490010
// MI455X (gfx1250) — compile-verified
//
#include <hip/hip_runtime.h>

// Problem shape (fixed by the harness/reference).
#define B_  16
#define N_  8
#define C_  3
#define H_  256
#define W_  256
#define HW_ (H_ * W_)
#define THREADS 1024

// Vector + address-space types matching the gfx1250 async-LDS builtin signature:
//   void __builtin_amdgcn_global_load_async_to_lds_b128(int4 AS(1)* src,
//                                                       int4 AS(3)* dst,
//                                                       imm offset, imm cpol)
typedef int v4i_ __attribute__((vector_size(4 * sizeof(int))));
typedef __attribute__((address_space(1))) v4i_* gptr_v4i;
typedef __attribute__((address_space(3))) v4i_* lptr_v4i;

// ---------------------------------------------------------------------------
// Main warp kernel: one block per (b, n, c). Stage the (b,c) source plane
// (256 KB) into LDS via gfx1250 async global->LDS loads, then bilinear-sample
// it for the (b,n) inverse-affine transform.
// ---------------------------------------------------------------------------
__global__ __launch_bounds__(THREADS) void affine_warp_kernel(
    const float* __restrict__ img,     // (B, C, H, W)
    const float* __restrict__ xf,      // (B, N, 6)
    float* __restrict__ out,           // (B, N, C, H, W)
    int* __restrict__ flags)           // (B*N) "any nonzero" flags
{
    __shared__ float smem[HW_];        // 256 KB of the 320 KB WGP LDS
    __shared__ int   blockAny;

    const int blk = blockIdx.x;
    const int c   = blk % C_;
    const int n   = (blk / C_) % N_;
    const int b   = blk / (C_ * N_);
    const int tid = threadIdx.x;

    if (tid == 0) blockAny = 0;

    const float* plane = img + (size_t)(b * C_ + c) * HW_;

    // ---- Stage source plane into LDS (async CDNA5 path, ASYNCcnt) ---------
#if defined(__gfx1250__) && __has_builtin(__builtin_amdgcn_global_load_async_to_lds_b128)
    for (int i = tid * 4; i < HW_; i += THREADS * 4) {
        __builtin_amdgcn_global_load_async_to_lds_b128(
            (gptr_v4i)(plane + i),
            (lptr_v4i)(&smem[i]),
            /*offset=*/0, /*cpol=*/0);
    }
  #if __has_builtin(__builtin_amdgcn_s_wait_asynccnt)
    __builtin_amdgcn_s_wait_asynccnt(0);
  #else
    asm volatile("s_wait_asynccnt 0" ::: "memory");
  #endif
#else
    // Fallback: vectorized synchronous copy through VGPRs.
    for (int i = tid * 4; i < HW_; i += THREADS * 4) {
        *(float4*)(&smem[i]) = *(const float4*)(plane + i);
    }
#endif
    __syncthreads();

    // ---- Uniform per-(b,n) inverse affine ---------------------------------
    const float* t  = xf + (size_t)(b * N_ + n) * 6;
    const float A   = t[0], Bm = t[1], TX = t[2];
    const float Cm  = t[3], D  = t[4], TY = t[5];
    const float det = A * D - Bm * Cm;
    const float ia  =  D  / det;
    const float ib  = -Bm / det;
    const float ic  = -Cm / det;
    const float idv =  A  / det;

    float* outp = out + ((size_t)(b * N_ + n) * C_ + c) * HW_;

    int any = 0;
    for (int pix = tid; pix < HW_; pix += THREADS) {
        const int x = pix & (W_ - 1);
        const int y = pix >> 8;

        const float dx = (float)x - TX;
        const float dy = (float)y - TY;
        const float sx = ia * dx + ib * dy;
        const float sy = ic * dx + idv * dy;

        const float x0f = floorf(sx);
        const float y0f = floorf(sy);
        const float wx  = sx - x0f;
        const float wy  = sy - y0f;

        const int x0 = (int)x0f, y0 = (int)y0f;
        const int x1 = x0 + 1,   y1 = y0 + 1;

        const bool vx0 = (unsigned)x0 < (unsigned)W_;
        const bool vx1 = (unsigned)x1 < (unsigned)W_;
        const bool vy0 = (unsigned)y0 < (unsigned)H_;
        const bool vy1 = (unsigned)y1 < (unsigned)H_;

        const float v00 = (vx0 && vy0) ? smem[(y0 << 8) | x0] : 0.0f;
        const float v01 = (vx1 && vy0) ? smem[(y0 << 8) | x1] : 0.0f;
        const float v10 = (vx0 && vy1) ? smem[(y1 << 8) | x0] : 0.0f;
        const float v11 = (vx1 && vy1) ? smem[(y1 << 8) | x1] : 0.0f;

        const float r = v00 * (1.0f - wx) * (1.0f - wy)
                      + v01 * wx          * (1.0f - wy)
                      + v10 * (1.0f - wx) * wy
                      + v11 * wx          * wy;

        outp[pix] = r;
        any |= (r != 0.0f);
    }

    if (any) atomicOr(&blockAny, 1);
    __syncthreads();
    if (tid == 0 && blockAny) atomicOr(&flags[b * N_ + n], 1);
}

// ---------------------------------------------------------------------------
// Fixup: reference semantics — if an entire (b,n) slab (C,H,W) is all zeros,
// replace it with -1.0. Early-exits (one scalar load) in the common case.
// ---------------------------------------------------------------------------
__global__ __launch_bounds__(256) void allzero_fixup_kernel(
    float* __restrict__ out, const int* __restrict__ flags)
{
    const int bn = blockIdx.x;
    if (flags[bn] != 0) return;
    float* p = out + (size_t)bn * (C_ * HW_);
    for (int i = threadIdx.x; i < C_ * HW_; i += 256) {
        p[i] = -1.0f;
    }
}

extern "C" void kernel_launch(void* const* d_in, const int* in_sizes, int n_in,
                              void* d_out, int out_size, void* d_ws, size_t ws_size,
                              hipStream_t stream) {
    (void)in_sizes; (void)n_in; (void)out_size; (void)ws_size;
    const float* img = (const float*)d_in[0];  // (16,3,256,256) f32
    const float* xf  = (const float*)d_in[1];  // (16,8,6)       f32
    float* out = (float*)d_out;                // (16,8,3,256,256) f32
    int* flags = (int*)d_ws;                   // B*N ints

    (void)hipMemsetAsync(flags, 0, B_ * N_ * sizeof(int), stream);

    affine_warp_kernel<<<dim3(B_ * N_ * C_), dim3(THREADS), 0, stream>>>(
        img, xf, out, flags);

    allzero_fixup_kernel<<<dim3(B_ * N_), dim3(256), 0, stream>>>(out, flags);
}